// EnGMF_27565100106163
// MI455X (gfx1250) — compile-verified
//
#include <hip/hip_runtime.h>
#include <math.h>

typedef __attribute__((ext_vector_type(2))) float v2f;
typedef __attribute__((ext_vector_type(8))) float v8f;

#define COV_BLOCKS 128
#define WAVES_PER_BLOCK 8
#define TM 256  // ensemble members per LDS tile (TM*8 floats = 8KB per buffer)

// ---------------- ordered-float encoding for atomicMax on floats ----------------
__device__ inline unsigned encodeFloat(float f) {
  unsigned u = __float_as_uint(f);
  return (u & 0x80000000u) ? ~u : (u | 0x80000000u);
}
__device__ inline float decodeFloat(unsigned e) {
  unsigned v = (e & 0x80000000u) ? (e & 0x7FFFFFFFu) : ~e;
  return __uint_as_float(v);
}

// ---------------- threefry2x32 counter-based PRNG ----------------
__device__ inline unsigned rotl32(unsigned x, int r) { return (x << r) | (x >> (32 - r)); }

__device__ inline void threefry2x32(unsigned k0, unsigned k1, unsigned c0, unsigned c1,
                                    unsigned& o0, unsigned& o1) {
  unsigned ks2 = k0 ^ k1 ^ 0x1BD11BDAu;
  unsigned x0 = c0 + k0, x1 = c1 + k1;
#define TF_R4(a, b, c, d)                                   \
  x0 += x1; x1 = rotl32(x1, a); x1 ^= x0;                   \
  x0 += x1; x1 = rotl32(x1, b); x1 ^= x0;                   \
  x0 += x1; x1 = rotl32(x1, c); x1 ^= x0;                   \
  x0 += x1; x1 = rotl32(x1, d); x1 ^= x0;
  TF_R4(13, 15, 26, 6)  x0 += k1;  x1 += ks2 + 1u;
  TF_R4(17, 29, 16, 24) x0 += ks2; x1 += k0 + 2u;
  TF_R4(13, 15, 26, 6)  x0 += k0;  x1 += k1 + 3u;
  TF_R4(17, 29, 16, 24) x0 += k1;  x1 += ks2 + 4u;
  TF_R4(13, 15, 26, 6)  x0 += ks2; x1 += k0 + 5u;
#undef TF_R4
  o0 = x0; o1 = x1;
}

__device__ inline float u32_to_unit(unsigned b) {
  return __uint_as_float(0x3F800000u | (b >> 9)) - 1.0f;  // [0,1)
}

// ---------------- ws layout init ----------------
__global__ void init_kernel(float* wsf, unsigned* maxSlot) {
  int t = threadIdx.x;
  if (t < 72) wsf[t] = 0.0f;   // Sx[8] + Sxx[64]
  if (t == 0) *maxSlot = 0u;   // minimum of ordered encoding
}

// ---------------- Kernel 1: moments via async-LDS staging + WMMA ----------------
// Pipeline per block: GLOBAL_LOAD_ASYNC_TO_LDS_B128 stages a 256-member tile
// (8KB) into LDS (double buffered), s_wait_asynccnt + barrier, then each wave
// runs a 4x-unrolled chain of V_WMMA_F32_16X16X4_F32 accumulating two 8x8
// outer-product streams (two groups of 4 members packed into one 16x16 tile).
__global__ void __launch_bounds__(256) cov_wmma_kernel(const float* __restrict__ state,
                                                       float* __restrict__ Sx,
                                                       float* __restrict__ Sxx, int N) {
  __shared__ float tile[2][TM * 8];
  __shared__ float sSxx[64];
  __shared__ float sSx[8];
  int tid = threadIdx.x;
  if (tid < 64) sSxx[tid] = 0.0f;
  if (tid < 8) sSx[tid] = 0.0f;

  int lane = tid & 31;
  int wave = tid >> 5;

  // A-layout (32-bit 16x4): lanes 0-15 hold K=0 (vgpr0), K=1 (vgpr1);
  // lanes 16-31 hold K=2, K=3.
  int half = lane >> 4;   // selects K pair {0,1} vs {2,3}
  int r    = lane & 15;   // A row (M)
  int grp  = r >> 3;      // member group 0/1
  int dim  = r & 7;       // state dimension

  const char* gbase = (const char*)state;
  long long maxByte = (long long)N * 32 - 16;  // last valid 16B load
  int nTiles = (N + TM - 1) / TM;

  // Async-stage one tile (TM*32 bytes): 256 threads x 2 x 16B.
  auto stage = [&](int buf, int tileIdx) {
    long long tileByte = (long long)tileIdx * TM * 32;
#pragma unroll
    for (int p = 0; p < 2; p++) {
      long long off = tileByte + (long long)(p * 256 + tid) * 16;
      if (off > maxByte) off = maxByte;  // clamp tail (masked at compute)
      unsigned ldsOff =
          (unsigned)(uintptr_t)(&tile[buf][0]) + (unsigned)(p * 4096 + tid * 16);
      const char* g = gbase + off;
      asm volatile("global_load_async_to_lds_b128 %0, %1, off"
                   :
                   : "v"(ldsOff), "v"(g)
                   : "memory");
    }
  };

  v8f acc = {};
  float sxAcc = 0.0f;
  int buf = 0;
  int t0 = blockIdx.x;
  if (t0 < nTiles) stage(0, t0);

  for (int t = t0; t < nTiles; t += gridDim.x) {
    asm volatile("s_wait_asynccnt 0x0" ::: "memory");
    __syncthreads();  // tile[buf] staged by all waves

    int nxt = t + gridDim.x;
    if (nxt < nTiles) stage(buf ^ 1, nxt);

    int tileBase = t * TM;
    int lmBase = wave * 32 + grp * 4 + half * 2;  // local member for K pair
#pragma unroll
    for (int g = 0; g < 4; g++) {
      int lm0 = lmBase + g * 8;
      int m0 = tileBase + lm0;
      v2f a;
      a.x = (m0 < N) ? tile[buf][lm0 * 8 + dim] : 0.0f;
      a.y = (m0 + 1 < N) ? tile[buf][(lm0 + 1) * 8 + dim] : 0.0f;
      sxAcc += a.x + a.y;
      acc = __builtin_amdgcn_wmma_f32_16x16x4_f32(false, a, false, a, (short)0, acc,
                                                  false, false);
    }
    __syncthreads();  // compute done before tile[buf] is restaged next round
    buf ^= 1;
  }

  // C layout: vgpr i, lanes 0-15 -> C[M=i][N=lane]; lanes 16-31 -> C[M=8+i][N=lane-16].
  // Group0 partial: C[d0][d1] -> lanes 0-7.  Group1 partial: C[8+d0][8+d1] -> lanes 24-31.
  if (lane < 8) {
#pragma unroll
    for (int i = 0; i < 8; i++) atomicAdd(&sSxx[i * 8 + lane], acc[i]);
  } else if (lane >= 24) {
    int d1 = lane - 24;
#pragma unroll
    for (int i = 0; i < 8; i++) atomicAdd(&sSxx[i * 8 + d1], acc[i]);
  }
  atomicAdd(&sSx[dim], sxAcc);
  __syncthreads();
  if (tid < 64) atomicAdd(&Sxx[tid], sSxx[tid]);
  if (tid < 8) atomicAdd(&Sx[tid], sSx[tid]);
}

// ---------------- Kernel 2: finalize P = bw*cov and Cholesky L ----------------
__global__ void finalize_kernel(const float* Sx, const float* Sxx, const float* bandwidth,
                                float* Lout, int N) {
  if (threadIdx.x != 0 || blockIdx.x != 0) return;
  float bw = bandwidth[0];
  float invN = 1.0f / (float)N;
  float invNm1 = 1.0f / (float)(N - 1);
  float P[64];
  for (int i = 0; i < 8; i++)
    for (int j = 0; j < 8; j++)
      P[i * 8 + j] = bw * (Sxx[i * 8 + j] - Sx[i] * Sx[j] * invN) * invNm1;
  float L[64];
  for (int i = 0; i < 64; i++) L[i] = 0.0f;
  for (int c = 0; c < 8; c++) {
    float d = P[c * 8 + c];
    for (int k = 0; k < c; k++) d -= L[c * 8 + k] * L[c * 8 + k];
    float lcc = sqrtf(fmaxf(d, 1e-30f));
    L[c * 8 + c] = lcc;
    float inv = 1.0f / lcc;
    for (int rr = c + 1; rr < 8; rr++) {
      float s = P[rr * 8 + c];
      for (int k = 0; k < c; k++) s -= L[rr * 8 + k] * L[c * 8 + k];
      L[rr * 8 + c] = s * inv;
    }
  }
  for (int i = 0; i < 64; i++) Lout[i] = L[i];
}

// ---------------- shared per-member update (recomputed where needed) ----------------
__device__ inline void member_update(const float* __restrict__ state,
                                     const float* __restrict__ L, int j, float z, float R,
                                     float* x, float* uvec, float& uu, float& S, float* xn,
                                     float& norm) {
  const float4* s4 = reinterpret_cast<const float4*>(state) + (size_t)j * 2;
  float4 A = s4[0], B = s4[1];
  x[0] = A.x; x[1] = A.y; x[2] = A.z; x[3] = A.w;
  x[4] = B.x; x[5] = B.y; x[6] = B.z; x[7] = B.w;
  float n2 = 0.0f;
#pragma unroll
  for (int d = 0; d < 8; d++) n2 += x[d] * x[d];
  norm = sqrtf(n2);
  float invn = 1.0f / norm;
  float h[8];
#pragma unroll
  for (int d = 0; d < 8; d++) h[d] = x[d] * invn;
  // u = L^T h  (L lower triangular)
  uu = 0.0f;
#pragma unroll
  for (int d = 0; d < 8; d++) {
    float s = 0.0f;
    for (int k = d; k < 8; k++) s += L[k * 8 + d] * h[k];
    uvec[d] = s;
    uu += s * s;
  }
  S = uu + R;  // h^T P h + R
  float invS = 1.0f / S;
  float resid = norm - z;
  // x_new = x - (P h / S) * (norm - z),  P h = L u
#pragma unroll
  for (int d = 0; d < 8; d++) {
    float s = 0.0f;
    for (int k = 0; k <= d; k++) s += L[d * 8 + k] * uvec[k];
    xn[d] = x[d] - s * invS * resid;
  }
}

// ---------------- Kernel 3: log-weights + global max ----------------
__global__ void __launch_bounds__(256) logw_kernel(const float* __restrict__ state,
                                                   const float* __restrict__ Lmat,
                                                   const float* meas, const float* meas_cov,
                                                   float* __restrict__ logw,
                                                   unsigned* maxSlot, int N) {
  __shared__ float sL[64];
  __shared__ unsigned smax;
  if (threadIdx.x == 0) smax = 0u;
  if (threadIdx.x < 64) sL[threadIdx.x] = Lmat[threadIdx.x];
  __syncthreads();
  int i = blockIdx.x * 256 + threadIdx.x;
  if (i < N) {
    float z = meas[0], R = meas_cov[0];
    float x[8], uvec[8], xn[8], uu, S, norm;
    member_update(state, sL, i, z, R, x, uvec, uu, S, xn, norm);
    float hn2 = 0.0f;
#pragma unroll
    for (int d = 0; d < 8; d++) hn2 += xn[d] * xn[d];
    float hn = sqrtf(hn2);
    float dz = z - hn;
    float lw = -0.5f * (__logf(6.28318530718f * S) + dz * dz / S);
    logw[i] = lw;
    atomicMax(&smax, encodeFloat(lw));
  }
  __syncthreads();
  if (threadIdx.x == 0) atomicMax(maxSlot, smax);
}

// ---------------- Kernels 4a/4b/4c: exp + inclusive scan -> CDF ----------------
__global__ void __launch_bounds__(256) scan_local_kernel(const float* __restrict__ logw,
                                                         float* __restrict__ cdf,
                                                         float* __restrict__ bsum,
                                                         const unsigned* maxSlot, int N) {
  __shared__ float s[256];
  int t = threadIdx.x;
  int i = blockIdx.x * 256 + t;
  float mx = decodeFloat(*maxSlot);
  float w = (i < N) ? __expf(logw[i] - mx) : 0.0f;
  s[t] = w;
  __syncthreads();
  for (int off = 1; off < 256; off <<= 1) {
    float prev = (t >= off) ? s[t - off] : 0.0f;
    __syncthreads();
    s[t] += prev;
    __syncthreads();
  }
  if (i < N) cdf[i] = s[t];
  if (t == 255) bsum[blockIdx.x] = s[255];
}

__global__ void __launch_bounds__(1024) scan_bsums_kernel(float* bsum, int nb) {
  __shared__ float s[1024];
  int t = threadIdx.x;
  s[t] = (t < nb) ? bsum[t] : 0.0f;
  __syncthreads();
  for (int off = 1; off < 1024; off <<= 1) {
    float prev = (t >= off) ? s[t - off] : 0.0f;
    __syncthreads();
    s[t] += prev;
    __syncthreads();
  }
  if (t < nb) bsum[t] = s[t];
}

__global__ void __launch_bounds__(256) scan_addoff_kernel(float* __restrict__ cdf,
                                                          const float* __restrict__ bsum,
                                                          int N) {
  int i = blockIdx.x * 256 + threadIdx.x;
  if (i < N && blockIdx.x > 0) cdf[i] += bsum[blockIdx.x - 1];
}

// ---------------- Kernel 5: resample + rank-1-sqrt Gaussian sampling ----------------
__global__ void __launch_bounds__(256) sample_kernel(const float* __restrict__ state,
                                                     const float* __restrict__ Lmat,
                                                     const float* __restrict__ cdf,
                                                     const float* meas, const float* meas_cov,
                                                     float* __restrict__ out, int N) {
  __shared__ float sL[64];
  if (threadIdx.x < 64) sL[threadIdx.x] = Lmat[threadIdx.x];
  __syncthreads();
  int i = blockIdx.x * 256 + threadIdx.x;
  if (i >= N) return;

  float total = cdf[N - 1];

  // inverse-CDF categorical draw
  unsigned r0, r1;
  threefry2x32(0x2A2A2A2Au, 42u, (unsigned)i, 1u, r0, r1);
  float t = u32_to_unit(r0) * total;
  int lo = 0, hi = N - 1;
  while (lo < hi) {
    int mid = (lo + hi) >> 1;
    if (cdf[mid] < t) lo = mid + 1; else hi = mid;
  }
  int j = lo;

  float z = meas[0], R = meas_cov[0];
  float x[8], uvec[8], xn[8], uu, S, norm;
  member_update(state, sL, j, z, R, x, uvec, uu, S, xn, norm);

  // cov_post^(1/2) = L (I - alpha * u u^T),  alpha = (1 - sqrt(R/S)) / |u|^2
  float alpha = (1.0f - sqrtf(fmaxf(R, 0.0f) / S)) / fmaxf(uu, 1e-30f);

  // 8 standard normals via Box-Muller
  float g[8];
#pragma unroll
  for (int p = 0; p < 4; p++) {
    unsigned a, b;
    threefry2x32(0x2A2A2A2Au, 42u, (unsigned)i, 2u + (unsigned)p, a, b);
    float u1 = fmaxf(u32_to_unit(a), 1e-12f);
    float u2 = u32_to_unit(b);
    float rad = sqrtf(-2.0f * __logf(u1));
    float sn, cs;
    __sincosf(6.28318530718f * u2, &sn, &cs);
    g[2 * p] = rad * cs;
    g[2 * p + 1] = rad * sn;
  }
  float ug = 0.0f;
#pragma unroll
  for (int d = 0; d < 8; d++) ug += uvec[d] * g[d];
  float y[8];
#pragma unroll
  for (int d = 0; d < 8; d++) y[d] = g[d] - alpha * ug * uvec[d];

  float samp[8];
#pragma unroll
  for (int d = 0; d < 8; d++) {
    float s = 0.0f;
    for (int k = 0; k <= d; k++) s += sL[d * 8 + k] * y[k];
    samp[d] = xn[d] + s;
  }
  float4* o4 = reinterpret_cast<float4*>(out) + (size_t)i * 2;
  o4[0] = make_float4(samp[0], samp[1], samp[2], samp[3]);
  o4[1] = make_float4(samp[4], samp[5], samp[6], samp[7]);
}

// ---------------- host-side launcher ----------------
extern "C" void kernel_launch(void* const* d_in, const int* in_sizes, int n_in,
                              void* d_out, int out_size, void* d_ws, size_t ws_size,
                              hipStream_t stream) {
  const float* state     = (const float*)d_in[0];
  const float* meas      = (const float*)d_in[1];
  const float* bandwidth = (const float*)d_in[2];
  const float* meas_cov  = (const float*)d_in[3];
  float* out = (float*)d_out;
  int N = in_sizes[0] / 8;

  float* wsf = (float*)d_ws;
  float* Sx   = wsf;            // 8
  float* Sxx  = wsf + 8;        // 64
  float* Lmat = wsf + 72;       // 64
  unsigned* maxSlot = (unsigned*)(wsf + 136);
  float* logw = wsf + 256;      // N
  float* cdf  = logw + N;       // N
  float* bsum = cdf + N;        // nb (<= 1024)
  int nb = (N + 255) / 256;

  init_kernel<<<1, 256, 0, stream>>>(wsf, maxSlot);
  cov_wmma_kernel<<<COV_BLOCKS, 256, 0, stream>>>(state, Sx, Sxx, N);
  finalize_kernel<<<1, 1, 0, stream>>>(Sx, Sxx, bandwidth, Lmat, N);
  logw_kernel<<<nb, 256, 0, stream>>>(state, Lmat, meas, meas_cov, logw, maxSlot, N);
  scan_local_kernel<<<nb, 256, 0, stream>>>(logw, cdf, bsum, maxSlot, N);
  scan_bsums_kernel<<<1, 1024, 0, stream>>>(bsum, nb);
  scan_addoff_kernel<<<nb, 256, 0, stream>>>(cdf, bsum, N);
  sample_kernel<<<nb, 256, 0, stream>>>(state, Lmat, cdf, meas, meas_cov, out, N);
}